// DecentLayer_11613591569241
// MI455X (gfx1250) — compile-verified
//
#include <hip/hip_runtime.h>
#include <stdint.h>

// fp32 3x3 conv (implicit GEMM) via V_WMMA_F32_16X16X4_F32 on gfx1250,
// with GLOBAL_LOAD_ASYNC_TO_LDS_B128 staging of the input tile.
// out(16,32,256,256) = conv(x gathered to 16 ch, w(32,16,3,3)), pad 1, stride 1.

typedef __attribute__((ext_vector_type(2))) float v2f;
typedef __attribute__((ext_vector_type(4))) float v4f;
typedef __attribute__((ext_vector_type(8))) float v8f;

#define NF     32                 // output filters
#define CX     24                 // x channels
#define CIN    16                 // gathered input channels
#define BATCH  16
#define HH     256
#define WW     256
#define WT     64                 // output w-positions per block
#define HB     8                  // output rows per block
#define XROWS  (HB + 2)           // 10 input rows (halo)
#define XW     80                 // padded row pitch: halo wl=3 / wl=68, interior 4..67 (16B aligned)
#define CH_STRIDE (XROWS * XW + 16) // 816 dwords: half-wave bank de-conflict
#define KTOT   (CIN * 9)          // 144
#define WPITCH 145                // 17-bank lane stride for A-fragment reads

__global__ __launch_bounds__(256)
void decent_conv_wmma(const float* __restrict__ x,
                      const int* __restrict__ ms_x, const int* __restrict__ ns_x,
                      const int* __restrict__ ms_in, const int* __restrict__ ns_in,
                      const float* __restrict__ wgt,
                      float* __restrict__ out)
{
    __shared__ __attribute__((aligned(16))) float s_x[CIN * CH_STRIDE]; // ~51 KB
    __shared__ float s_w[NF * WPITCH];                                  // ~18.6 KB
    __shared__ int   s_xid[CIN];

    const int tid    = threadIdx.x;
    const int wchunk = blockIdx.x;          // 0..3   (w strip)
    const int h0     = blockIdx.y * HB;     // first output row of this block
    const int b      = blockIdx.z;          // 0..15
    const int wbase  = wchunk * WT;

    // --- phase 0: channel gather map (each filter input ch -> unique x ch) ---
    if (tid < CIN) {
        const int mi = ms_in[tid], ni = ns_in[tid];
        int id = 0;
        for (int i = 0; i < CX; ++i)
            if (ms_x[i] == mi && ns_x[i] == ni) id = i;
        s_xid[tid] = id;
    }
    // --- phase 0b: weights -> LDS (4608 dwords, padded pitch) ---
    for (int i = tid; i < NF * KTOT; i += 256) {
        const int o = i / KTOT;
        const int r = i - o * KTOT;
        s_w[o * WPITCH + r] = wgt[i];
    }
    __syncthreads();   // s_xid visible before x staging

    // --- phase 1a: interior x tile -> LDS via async B128 copies ---
    // 16ch x 10 rows x 16 chunks of float4; interior gw = wbase..wbase+63 always in range
    for (int i = tid; i < CIN * XROWS * 16; i += 256) {
        const int c    = i / (XROWS * 16);
        const int rem  = i - c * (XROWS * 16);
        const int row  = rem >> 4;
        const int j    = rem & 15;
        const int gh   = h0 + row - 1;
        const int lidx = c * CH_STRIDE + row * XW + 4 + (j << 2);
        if ((unsigned)gh < (unsigned)HH) {
            const int ch = s_xid[c];
            const float* gp =
                &x[(((long)b * CX + ch) * HH + gh) * WW + wbase + (j << 2)];
            const unsigned lds_off = (unsigned)(uintptr_t)(&s_x[lidx]);
            const unsigned long long ga = (unsigned long long)(uintptr_t)gp;
            asm volatile("global_load_async_to_lds_b128 %0, %1, off"
                         :: "v"(lds_off), "v"(ga) : "memory");
        } else {
            v4f z = {};
            *(v4f*)&s_x[lidx] = z;   // zero pad rows outside the image
        }
    }
    // --- phase 1b: halo columns (gw = wbase-1 and wbase+64), scalar path ---
    for (int i = tid; i < CIN * XROWS * 2; i += 256) {
        const int c    = i / (XROWS * 2);
        const int rem  = i - c * (XROWS * 2);
        const int row  = rem >> 1;
        const int side = rem & 1;
        const int gh   = h0 + row - 1;
        const int gw   = wbase + (side ? WT : -1);
        const int wl   = side ? (4 + WT) : 3;
        float v = 0.0f;
        if ((unsigned)gh < (unsigned)HH && (unsigned)gw < (unsigned)WW)
            v = x[(((long)b * CX + s_xid[c]) * HH + gh) * WW + gw];
        s_x[c * CH_STRIDE + row * XW + wl] = v;
    }
    asm volatile("s_wait_asynccnt 0x0" ::: "memory");
    __syncthreads();

    // --- phase 2: 8 waves; wave owns 16(M filters) x 16(N w-pos), loops 8 rows ---
    const int wave  = tid >> 5;
    const int lane  = tid & 31;
    const int mbase = (wave & 1) * 16;      // filter tile base
    const int nb    = (wave >> 1) * 16;     // w tile base within strip
    const int l16   = lane & 15;
    const int koff  = (lane >> 4) << 1;     // halves hold K+0/1 vs K+2/3 (ISA layout)
    const int m0    = mbase + ((lane >> 4) << 3);
    const int gws   = wbase + nb + l16;

    for (int r = 0; r < HB; ++r) {
        v8f acc = {};
        #pragma unroll
        for (int kh = 0; kh < 3; ++kh) {
            #pragma unroll
            for (int kw = 0; kw < 3; ++kw) {
                const float* xp = &s_x[(r + kh) * XW + nb + l16 + kw + 3]; // + c*CH_STRIDE
                const float* wp = &s_w[(mbase + l16) * WPITCH + kh * 3 + kw]; // + c*9
                #pragma unroll
                for (int kb = 0; kb < CIN; kb += 4) {
                    const int c0 = kb + koff;
                    v2f a, bf;
                    a.x  = wp[c0 * 9];
                    a.y  = wp[(c0 + 1) * 9];
                    bf.x = xp[c0 * CH_STRIDE];
                    bf.y = xp[(c0 + 1) * CH_STRIDE];
                    // (neg_a, A, neg_b, B, c_mod, C, reuse_a, reuse_b)
                    acc = __builtin_amdgcn_wmma_f32_16x16x4_f32(
                            false, a, false, bf, (short)0, acc, false, false);
                }
            }
        }
        // store D per 16x16 f32 C/D layout: lane half selects M / M+8, col = l16
        const int h = h0 + r;
        #pragma unroll
        for (int v = 0; v < 8; ++v)
            out[(((long)b * NF + (m0 + v)) * HH + h) * WW + gws] = acc[v];
    }
}

extern "C" void kernel_launch(void* const* d_in, const int* in_sizes, int n_in,
                              void* d_out, int out_size, void* d_ws, size_t ws_size,
                              hipStream_t stream) {
    const float* x     = (const float*)d_in[0];
    const int*   ms_x  = (const int*)  d_in[1];
    const int*   ns_x  = (const int*)  d_in[2];
    const int*   ms_in = (const int*)  d_in[3];
    const int*   ns_in = (const int*)  d_in[4];
    const float* w     = (const float*)d_in[5];
    float*       out   = (float*)      d_out;

    dim3 grid(WW / WT, HH / HB, BATCH);   // 4 x 32 x 16 = 2048 workgroups
    decent_conv_wmma<<<grid, 256, 0, stream>>>(x, ms_x, ns_x, ms_in, ns_in, w, out);
}